// GNN_69329362092402
// MI455X (gfx1250) — compile-verified
//
#include <hip/hip_runtime.h>

typedef float v2f __attribute__((ext_vector_type(2)));
typedef float v8f __attribute__((ext_vector_type(8)));

#define NF 64          // padded feature stride
#define GEMM_BLOCK 256 // 8 waves

// ---------------- degree / normalization ----------------

__global__ void k_init_deg(float* deg, int n) {
    int i = blockIdx.x * blockDim.x + threadIdx.x;
    if (i < n) deg[i] = 1.0f;  // self-loop contributes 1 to every node's degree
}

__global__ void k_count_deg(const long long* __restrict__ dst, float* deg, int E) {
    int e = blockIdx.x * blockDim.x + threadIdx.x;
    if (e < E) atomicAdd(&deg[(int)dst[e]], 1.0f);
}

__global__ void k_deg_to_dinv(float* deg, int n) {
    int i = blockIdx.x * blockDim.x + threadIdx.x;
    if (i < n) {
        float d = deg[i];
        deg[i] = (d > 0.0f) ? rsqrtf(d) : 0.0f;
    }
}

__global__ void k_zero(float* p, long long n) {
    long long i = (long long)blockIdx.x * blockDim.x + threadIdx.x;
    long long stride = (long long)gridDim.x * blockDim.x;
    for (; i < n; i += stride) p[i] = 0.0f;
}

// ---------------- WMMA f32 GEMM: out[N x (ntiles*16), stride 64] = in[N x 64] @ W[64 x ncols] ----------------
// One wave computes a 16-row strip; 8 waves per block -> 128 rows per block.
__global__ __launch_bounds__(GEMM_BLOCK)
void k_gemm_wmma(const float* __restrict__ in, const float* __restrict__ W,
                 float* __restrict__ out, int n, int ncols, int ldw, int ntiles) {
    __shared__ float wlds[64 * 64];  // W staged in LDS, zero-padded to 64 cols
    int tid = threadIdx.x;
    for (int idx = tid; idx < 64 * 64; idx += GEMM_BLOCK) {
        int k = idx >> 6, c = idx & 63;
        wlds[idx] = (c < ncols) ? W[k * ldw + c] : 0.0f;
    }
    __syncthreads();

    int wave = tid >> 5;
    int lane = tid & 31;
    int ml   = lane & 15;          // M (for A) / N (for B) within tile
    int kb   = (lane >> 4) << 1;   // 0 for lanes 0-15, 2 for lanes 16-31
    int row0 = blockIdx.x * (GEMM_BLOCK / 32 * 16) + wave * 16;

    int arow = row0 + ml;
    if (arow > n - 1) arow = n - 1;            // clamp (keep EXEC all-1s for WMMA)
    const float* arowp = in + (long long)arow * NF;

    // A fragments per ISA 32-bit 16x4 layout: a.x = A[M][k+kb], a.y = A[M][k+kb+1]
    v2f a[16];
#pragma unroll
    for (int kk = 0; kk < 16; ++kk)
        a[kk] = *reinterpret_cast<const v2f*>(arowp + kk * 4 + kb);

    for (int t = 0; t < ntiles; ++t) {
        v8f c = {};
        int cbase = t * 16 + ml;
#pragma unroll
        for (int kk = 0; kk < 16; ++kk) {
            v2f b;
            b.x = wlds[(kk * 4 + kb) * 64 + cbase];
            b.y = wlds[(kk * 4 + kb + 1) * 64 + cbase];
            // D = A(16x4,f32) * B(4x16,f32) + C(16x16,f32)
            c = __builtin_amdgcn_wmma_f32_16x16x4_f32(
                    /*neg_a=*/false, a[kk], /*neg_b=*/false, b,
                    /*c_mod=*/(short)0, c, /*reuse_a=*/false, /*reuse_b=*/false);
        }
        // D layout: VGPR v -> M = row0 + v + (lane>=16 ? 8 : 0), N = lane&15
        int mbase = row0 + ((lane >> 4) << 3);
#pragma unroll
        for (int v = 0; v < 8; ++v) {
            int r = mbase + v;
            if (r < n) out[(long long)r * NF + cbase] = c[v];
        }
    }
}

// ---------------- edge scatter: agg[dst] += dinv[src]*dinv[dst] * h[src] ----------------
__global__ void k_scatter(const long long* __restrict__ src, const long long* __restrict__ dst,
                          const float* __restrict__ dinv, const float* __restrict__ h,
                          float* agg, int E, int nf) {
    long long t = (long long)blockIdx.x * blockDim.x + threadIdx.x;
    int e = (int)(t >> 6);
    int f = (int)(t & 63);
    if (e >= E || f >= nf) return;
    int s = (int)src[e];
    int d = (int)dst[e];
    float w = dinv[s] * dinv[d];
    atomicAdd(&agg[(long long)d * NF + f], w * h[(long long)s * NF + f]);
}

// ---------------- epilogues ----------------
// y = relu(agg + dinv^2 * h + b)   (adds the self-loop message)
__global__ void k_final_relu(const float* __restrict__ agg, const float* __restrict__ h,
                             const float* __restrict__ dinv, const float* __restrict__ b,
                             float* y, int n) {
    long long t = (long long)blockIdx.x * blockDim.x + threadIdx.x;
    int i = (int)(t >> 6);
    int f = (int)(t & 63);
    if (i >= n) return;
    float di = dinv[i];
    float v = agg[t] + di * di * h[t] + b[f];
    y[t] = v > 0.0f ? v : 0.0f;
}

// out (tight stride 37) = agg + dinv^2 * h + b
__global__ void k_final_out(const float* __restrict__ agg, const float* __restrict__ h,
                            const float* __restrict__ dinv, const float* __restrict__ b,
                            float* out, int n, int ncols) {
    long long t = (long long)blockIdx.x * blockDim.x + threadIdx.x;
    if (t >= (long long)n * ncols) return;
    int i = (int)(t / ncols);
    int f = (int)(t % ncols);
    long long p = (long long)i * NF + f;
    float di = dinv[i];
    out[t] = agg[p] + di * di * h[p] + b[f];
}

extern "C" void kernel_launch(void* const* d_in, const int* in_sizes, int n_in,
                              void* d_out, int out_size, void* d_ws, size_t ws_size,
                              hipStream_t stream) {
    const float*      x  = (const float*)d_in[0];
    const long long*  ei = (const long long*)d_in[1];
    const float*      W1 = (const float*)d_in[2];
    const float*      b1 = (const float*)d_in[3];
    const float*      W2 = (const float*)d_in[4];
    const float*      b2 = (const float*)d_in[5];

    const int n  = in_sizes[0] / 64;   // 50000 nodes
    const int E  = in_sizes[1] / 2;    // 800000 edges
    const int c2 = in_sizes[4] / 64;   // 37 output cols

    const long long* src = ei;
    const long long* dst = ei + E;

    float* ws   = (float*)d_ws;
    long long npad = ((long long)n + 63) & ~63LL;
    float* dinv = ws;                          // n floats (deg -> dinv in place)
    float* h    = ws + npad;                   // n*64
    float* agg  = h + (long long)n * NF;       // n*64
    float* y    = agg + (long long)n * NF;     // n*64

    const long long nNF = (long long)n * NF;
    int g256n  = (n + 255) / 256;
    int g256E  = (E + 255) / 256;
    int gEdgeF = (int)(((long long)E * 64 + 255) / 256);
    int gNodeF = (int)((nNF + 255) / 256);

    // degree + symmetric normalization
    k_init_deg<<<g256n, 256, 0, stream>>>(dinv, n);
    k_count_deg<<<g256E, 256, 0, stream>>>(dst, dinv, E);
    k_deg_to_dinv<<<g256n, 256, 0, stream>>>(dinv, n);

    // layer 1: h = x @ W1 ; agg = scatter ; y = relu(agg + self + b1)
    k_gemm_wmma<<<(n + 127) / 128, GEMM_BLOCK, 0, stream>>>(x, W1, h, n, 64, 64, 4);
    k_zero<<<2048, 256, 0, stream>>>(agg, nNF);
    k_scatter<<<gEdgeF, 256, 0, stream>>>(src, dst, dinv, h, agg, E, 64);
    k_final_relu<<<gNodeF, 256, 0, stream>>>(agg, h, dinv, b1, y, n);

    // layer 2: h = y @ W2 (37 cols, padded to 48 computed / 64 stored stride)
    k_gemm_wmma<<<(n + 127) / 128, GEMM_BLOCK, 0, stream>>>(y, W2, h, n, c2, c2, 3);
    k_zero<<<2048, 256, 0, stream>>>(agg, nNF);
    k_scatter<<<gEdgeF, 256, 0, stream>>>(src, dst, dinv, h, agg, E, c2);

    long long nOut = (long long)n * c2;
    k_final_out<<<(int)((nOut + 255) / 256), 256, 0, stream>>>(agg, h, dinv, b2,
                                                              (float*)d_out, n, c2);
}